// Network_64201171140958
// MI455X (gfx1250) — compile-verified
//
#include <hip/hip_runtime.h>
#include <math.h>

// =====================================================================
// PointNet++ (SSG) forward + CVAE head for MI455X (gfx1250, wave32).
//
// Input flattening assumption (setup_inputs dict insertion order, params
// tree flattened in insertion order, tuples in order):
//   0: pcs (8,8192,4)  1: cp (8,3)  2: task (8,3)  3: noise (8,128)
//   4..39  : sa[l][conv j] = (W,g,b)  l=0..3, j=0..2   -> base 4+9l+3j
//   40..66 : fp[0] 3 convs (40,43,46), fp[1] 2 (49,52), fp[2] 2 (55,58),
//            fp[3] 2 (61,64); each conv = (W,g,b)
//   67..69 : fc (W,g,b)
//   70,71  : mlp_cp (W,b)    72,73: mlp_task (W,b)
//   74..83 : enc mlp1,mlp2,mlp3,mu,logvar (W,b each)
//   84..87 : dec l1 (W,b), l2 (W,b)
// =====================================================================

typedef float v2f __attribute__((ext_vector_type(2)));
typedef float v8f __attribute__((ext_vector_type(8)));

#define BN_EPS 1e-5f

// ---------------------------------------------------------------------
// Generic GEMM: Y[M x C] = X[M x K] * W[C x K]^T  via V_WMMA_F32_16X16X4_F32
// Requires M % 64 == 0, C % 32 == 0 (true for every layer here); K ragged.
// Block = 256 threads (8 waves). Tile 64(M) x 32(N); wave grid 4x2.
// Staging is register-buffered and software-pipelined: global loads for
// chunk k+1 issue before the WMMAs of chunk k, so the 6 independent loads
// per thread overlap the matrix work (no per-element load->wait->store
// chains). Prefetch reaches 2 chunks ahead (global_prefetch_b8).
// ---------------------------------------------------------------------
__global__ __launch_bounds__(256) void gemm_xwT_kernel(
    const float* __restrict__ X, const float* __restrict__ W,
    float* __restrict__ Y, int M, int K, int C) {
  __shared__ float Xs[64][17];
  __shared__ float Ws[32][17];
  const int tid  = threadIdx.x;
  const int lane = tid & 31;
  const int wave = tid >> 5;
  const int waveM = wave >> 1;        // 0..3
  const int waveN = wave & 1;         // 0..1
  const int tileM = blockIdx.x * 64;
  const int tileN = blockIdx.y * 32;

  // Fixed per-thread staging coordinates: column kk, base row (tid>>4),
  // X covers rows +0/+16/+32/+48, W covers rows +0/+16.
  const int kk   = tid & 15;
  const int srow = tid >> 4;          // 0..15
  float xv[4], wv[2];

  auto loadChunk = [&](int k0) {
    const int gk = k0 + kk;
    const bool kin = gk < K;
#pragma unroll
    for (int u = 0; u < 4; ++u)
      xv[u] = kin ? X[(size_t)(tileM + srow + 16 * u) * K + gk] : 0.0f;
#pragma unroll
    for (int u = 0; u < 2; ++u)
      wv[u] = kin ? W[(size_t)(tileN + srow + 16 * u) * K + gk] : 0.0f;
  };
  auto storeChunk = [&]() {
#pragma unroll
    for (int u = 0; u < 4; ++u) Xs[srow + 16 * u][kk] = xv[u];
#pragma unroll
    for (int u = 0; u < 2; ++u) Ws[srow + 16 * u][kk] = wv[u];
  };

  v8f acc = {};
  const int m  = waveM * 16 + (lane & 15);
  const int n  = waveN * 16 + (lane & 15);
  const int h2 = (lane >> 4) << 1;    // 0 or 2 (K-pair select per half-wave)

  loadChunk(0);
  storeChunk();
  __syncthreads();
  for (int k0 = 0;;) {
    const int kn = k0 + 16;
    const bool more = kn < K;          // uniform across block
    if (more) {
      loadChunk(kn);                   // issue early; waited at storeChunk
      if (kn + 16 < K) {               // prefetch 2 chunks ahead
        __builtin_prefetch(&X[(size_t)(tileM + srow) * K + kn + 16 + kk], 0, 1);
        __builtin_prefetch(&W[(size_t)(tileN + srow) * K + kn + 16 + kk], 0, 1);
      }
    }
#pragma unroll
    for (int j = 0; j < 4; ++j) {
      const int ks = j * 4 + h2;
      v2f a; a.x = Xs[m][ks]; a.y = Xs[m][ks + 1];   // A: 16x4 fp32 frag
      v2f b; b.x = Ws[n][ks]; b.y = Ws[n][ks + 1];   // B: 4x16 (= W^T tile)
      acc = __builtin_amdgcn_wmma_f32_16x16x4_f32(
          false, a, false, b, (short)0, acc, false, false);
    }
    if (!more) break;
    __syncthreads();                   // all waves done reading LDS
    storeChunk();
    __syncthreads();
    k0 = kn;
  }
  const int nOut  = tileN + waveN * 16 + (lane & 15);
  const int mBase = tileM + waveM * 16 + ((lane >> 4) << 3);
#pragma unroll
  for (int v = 0; v < 8; ++v)
    Y[(size_t)(mBase + v) * C + nOut] = acc[v];
}

// ---------------------------------------------------------------------
// Prep: apply cp substitution at row 0; split xyz (3ch) / feats (4ch).
// ---------------------------------------------------------------------
__global__ void prep_kernel(const float* __restrict__ pcs,
                            const float* __restrict__ cp,
                            float* __restrict__ xyz, float* __restrict__ feats,
                            int B, int N) {
  size_t total = (size_t)B * N;
  for (size_t t = blockIdx.x * (size_t)blockDim.x + threadIdx.x; t < total;
       t += (size_t)gridDim.x * blockDim.x) {
    int n = (int)(t % N), b = (int)(t / N);
    float x, y, z, w;
    if (n == 0) { x = cp[b*3]; y = cp[b*3+1]; z = cp[b*3+2]; w = 1.0f; }
    else {
      const float* p = pcs + t * 4;
      x = p[0]; y = p[1]; z = p[2]; w = p[3];
    }
    xyz[t*3] = x; xyz[t*3+1] = y; xyz[t*3+2] = z;
    feats[t*4] = x; feats[t*4+1] = y; feats[t*4+2] = z; feats[t*4+3] = w;
  }
}

// ---------------------------------------------------------------------
// Farthest point sampling: one 1024-thread block per batch. Points and the
// running min-distance stay in registers; each serial step = broadcast +
// wave32 shuffle reduce + cross-wave LDS reduce. Tie-break: lowest index.
// ---------------------------------------------------------------------
__global__ __launch_bounds__(1024) void fps_kernel(
    const float* __restrict__ xyz, int* __restrict__ fidx, int N, int npoint) {
  const int b = blockIdx.x, tid = threadIdx.x;
  const float* P = xyz + (size_t)b * N * 3;
  float px[8], py[8], pz[8], mind[8];
  int cnt = 0;
  for (int i = tid; i < N; i += 1024) {
    px[cnt] = P[i*3]; py[cnt] = P[i*3+1]; pz[cnt] = P[i*3+2];
    mind[cnt] = 1e10f; ++cnt;
  }
  __shared__ float s_last[3];
  __shared__ float s_val[32];
  __shared__ int   s_idx[32];
  __shared__ int   s_next;
  if (tid == 0) { fidx[(size_t)b*npoint] = 0; s_next = 0; }
  __syncthreads();
  for (int it = 1; it < npoint; ++it) {
    if (tid == 0) {
      int last = s_next;
      s_last[0] = P[last*3]; s_last[1] = P[last*3+1]; s_last[2] = P[last*3+2];
    }
    __syncthreads();
    const float lx = s_last[0], ly = s_last[1], lz = s_last[2];
    float bestv = -1.0f; int besti = N;
    int c = 0;
    for (int i = tid; i < N; i += 1024) {
      float dx = px[c]-lx, dy = py[c]-ly, dz = pz[c]-lz;
      float md = fminf(mind[c], dx*dx + dy*dy + dz*dz);
      mind[c] = md;
      if (md > bestv) { bestv = md; besti = i; }
      ++c;
    }
    for (int off = 16; off > 0; off >>= 1) {
      float ov = __shfl_xor(bestv, off, 32);
      int   oi = __shfl_xor(besti, off, 32);
      if (ov > bestv || (ov == bestv && oi < besti)) { bestv = ov; besti = oi; }
    }
    if ((tid & 31) == 0) { s_val[tid >> 5] = bestv; s_idx[tid >> 5] = besti; }
    __syncthreads();
    if (tid == 0) {
      float bv = s_val[0]; int bi = s_idx[0];
      for (int w = 1; w < 32; ++w)
        if (s_val[w] > bv || (s_val[w] == bv && s_idx[w] < bi)) { bv = s_val[w]; bi = s_idx[w]; }
      s_next = bi;
      fidx[(size_t)b*npoint + it] = bi;
    }
    __syncthreads();
  }
}

__global__ void gather_xyz_kernel(const float* __restrict__ xyz,
                                  const int* __restrict__ fidx,
                                  float* __restrict__ out, int B, int N, int S) {
  size_t total = (size_t)B * S * 3;
  for (size_t e = blockIdx.x * (size_t)blockDim.x + threadIdx.x; e < total;
       e += (size_t)gridDim.x * blockDim.x) {
    int c = (int)(e % 3); size_t bs = e / 3;
    int s = (int)(bs % S), b = (int)(bs / S);
    int g = fidx[(size_t)b*S + s];
    out[e] = xyz[((size_t)b*N + g)*3 + c];
  }
}

// Ball query: serial ascending scan (matches top_k-of-score ordering), pad
// with first in-range index.
__global__ void ball_query_kernel(const float* __restrict__ centers,
                                  const float* __restrict__ xyz,
                                  int* __restrict__ gidx,
                                  int B, int S, int N, float r2, int ns) {
  int t = blockIdx.x * blockDim.x + threadIdx.x;
  if (t >= B * S) return;
  int b = t / S;
  const float* c = centers + (size_t)t * 3;
  const float cx = c[0], cy = c[1], cz = c[2];
  const float* P = xyz + (size_t)b * N * 3;
  int* out = gidx + (size_t)t * ns;
  int cnt = 0, first = 0;
  for (int j = 0; j < N && cnt < ns; ++j) {
    float dx = P[j*3]-cx, dy = P[j*3+1]-cy, dz = P[j*3+2]-cz;
    if (dx*dx + dy*dy + dz*dz < r2) { if (cnt == 0) first = j; out[cnt++] = j; }
  }
  for (; cnt < ns; ++cnt) out[cnt] = first;
}

// Build grouped GEMM input: rows = B*S*ns, cols = 3 + Cf.
__global__ void group_kernel(const float* __restrict__ xyz,
                             const float* __restrict__ feats,
                             const float* __restrict__ centers,
                             const int* __restrict__ gidx,
                             float* __restrict__ X,
                             int B, int S, int N, int ns, int Cf) {
  const int Cin = 3 + Cf;
  size_t total = (size_t)B * S * ns * Cin;
  for (size_t e = blockIdx.x * (size_t)blockDim.x + threadIdx.x; e < total;
       e += (size_t)gridDim.x * blockDim.x) {
    int cc = (int)(e % Cin); size_t row = e / Cin;
    size_t bs = row / ns;
    int s = (int)(bs % S), b = (int)(bs / S);
    int g = gidx[row];
    float v;
    if (cc < 3)
      v = xyz[((size_t)b*N + g)*3 + cc] - centers[((size_t)b*S + s)*3 + cc];
    else
      v = feats[((size_t)b*N + g)*Cf + (cc - 3)];
    X[e] = v;
  }
}

// Per-channel training-mode BN stats -> fused (scale, shift).
__global__ __launch_bounds__(256) void bn_stats_kernel(
    const float* __restrict__ Y, const float* __restrict__ gamma,
    const float* __restrict__ beta, float* __restrict__ scale,
    float* __restrict__ shift, size_t M, int C) {
  const int c = blockIdx.x;
  float s = 0.0f, s2 = 0.0f;
  for (size_t i = threadIdx.x; i < M; i += 256) {
    float v = Y[i * C + c];
    s += v; s2 += v * v;
  }
  __shared__ float sh[256], sh2[256];
  sh[threadIdx.x] = s; sh2[threadIdx.x] = s2;
  __syncthreads();
  for (int o = 128; o > 0; o >>= 1) {
    if (threadIdx.x < (unsigned)o) {
      sh[threadIdx.x]  += sh[threadIdx.x + o];
      sh2[threadIdx.x] += sh2[threadIdx.x + o];
    }
    __syncthreads();
  }
  if (threadIdx.x == 0) {
    float mean = sh[0] / (float)M;
    float var  = sh2[0] / (float)M - mean * mean;
    float sc   = gamma[c] * rsqrtf(var + BN_EPS);
    scale[c] = sc;
    shift[c] = beta[c] - mean * sc;
  }
}

__global__ void bn_relu_kernel(float* __restrict__ Y,
                               const float* __restrict__ scale,
                               const float* __restrict__ shift,
                               size_t M, int C) {
  size_t total = M * (size_t)C;
  for (size_t e = blockIdx.x * (size_t)blockDim.x + threadIdx.x; e < total;
       e += (size_t)gridDim.x * blockDim.x) {
    int c = (int)(e % C);
    float v = Y[e] * scale[c] + shift[c];
    Y[e] = v > 0.0f ? v : 0.0f;
  }
}

__global__ void maxpool_kernel(const float* __restrict__ Y,
                               float* __restrict__ out,
                               int B, int S, int ns, int C) {
  size_t total = (size_t)B * S * C;
  for (size_t e = blockIdx.x * (size_t)blockDim.x + threadIdx.x; e < total;
       e += (size_t)gridDim.x * blockDim.x) {
    int c = (int)(e % C); size_t bs = e / C;
    float m = -3.4e38f;
    for (int s = 0; s < ns; ++s)
      m = fmaxf(m, Y[(bs * ns + s) * C + c]);
    out[e] = m;
  }
}

// 3-NN with first-occurrence tie-break + normalized inverse-distance weights.
__global__ void three_nn_kernel(const float* __restrict__ ux,
                                const float* __restrict__ kx,
                                int* __restrict__ idx3, float* __restrict__ w3,
                                int B, int Nu, int Nk) {
  int t = blockIdx.x * blockDim.x + threadIdx.x;
  if (t >= B * Nu) return;
  int b = t / Nu;
  const float* u = ux + (size_t)t * 3;
  const float x0 = u[0], y0 = u[1], z0 = u[2];
  const float* K = kx + (size_t)b * Nk * 3;
  float d0 = 1e30f, d1 = 1e30f, d2 = 1e30f;
  int i0 = 0, i1 = 0, i2 = 0;
  for (int m = 0; m < Nk; ++m) {
    float dx = K[m*3]-x0, dy = K[m*3+1]-y0, dz = K[m*3+2]-z0;
    float d = dx*dx + dy*dy + dz*dz;
    if (d < d0)      { d2=d1;i2=i1; d1=d0;i1=i0; d0=d;i0=m; }
    else if (d < d1) { d2=d1;i2=i1; d1=d;i1=m; }
    else if (d < d2) { d2=d;i2=m; }
  }
  float w0 = 1.0f/(d0 + 1e-8f), w1 = 1.0f/(d1 + 1e-8f), w2 = 1.0f/(d2 + 1e-8f);
  float ws = w0 + w1 + w2;
  idx3[t*3]=i0; idx3[t*3+1]=i1; idx3[t*3+2]=i2;
  w3[t*3]=w0/ws; w3[t*3+1]=w1/ws; w3[t*3+2]=w2/ws;
}

// Build FP GEMM input: [interp(kf @ 3NN weights) | uf], rows = B*Nu.
__global__ void fp_concat_kernel(const int* __restrict__ idx3,
                                 const float* __restrict__ w3,
                                 const float* __restrict__ kf,
                                 const float* __restrict__ uf,
                                 float* __restrict__ X,
                                 int B, int Nu, int Nk, int Ck, int Cu) {
  const int Cin = Ck + Cu;
  size_t total = (size_t)B * Nu * Cin;
  for (size_t e = blockIdx.x * (size_t)blockDim.x + threadIdx.x; e < total;
       e += (size_t)gridDim.x * blockDim.x) {
    int c = (int)(e % Cin); size_t row = e / Cin;
    int b = (int)(row / Nu);
    float v;
    if (c < Ck) {
      const float* kb = kf + (size_t)b * Nk * Ck;
      v = w3[row*3+0] * kb[(size_t)idx3[row*3+0]*Ck + c]
        + w3[row*3+1] * kb[(size_t)idx3[row*3+1]*Ck + c]
        + w3[row*3+2] * kb[(size_t)idx3[row*3+2]*Ck + c];
    } else {
      v = uf[row * (size_t)Cu + (c - Ck)];
    }
    X[e] = v;
  }
}

// ---------------------------------------------------------------------
// CVAE head: tiny MLPs, one block of 128 threads, serial over batch.
// ---------------------------------------------------------------------
__global__ __launch_bounds__(128) void head_kernel(
    const float* __restrict__ fcY, const float* __restrict__ fcScale,
    const float* __restrict__ fcShift,
    const float* __restrict__ cp, const float* __restrict__ task,
    const float* __restrict__ noise,
    const float* __restrict__ Wcp, const float* __restrict__ bcp,
    const float* __restrict__ Wtask, const float* __restrict__ btask,
    const float* __restrict__ W1, const float* __restrict__ b1,
    const float* __restrict__ W2, const float* __restrict__ b2,
    const float* __restrict__ W3, const float* __restrict__ b3,
    const float* __restrict__ Wmu, const float* __restrict__ bmu,
    const float* __restrict__ Wlv, const float* __restrict__ blv,
    const float* __restrict__ Wd1, const float* __restrict__ bd1,
    const float* __restrict__ Wd2, const float* __restrict__ bd2,
    float* __restrict__ out, int N) {
  __shared__ float net1[128], cpf[32], taskf[32], h[192];
  __shared__ float h1[128], h2[128], h3[128], zv[128], dvec[288], l1[128];
  const int tid = threadIdx.x;
  for (int b = 0; b < 8; ++b) {
    {
      float v = fcY[(size_t)b * N * 128 + tid] * fcScale[tid] + fcShift[tid];
      net1[tid] = v > 0.0f ? v : 0.0f;
    }
    if (tid < 32) {
      float s = bcp[tid], t2 = btask[tid];
      for (int k = 0; k < 3; ++k) {
        s  += Wcp[tid*3 + k]   * cp[b*3 + k];
        t2 += Wtask[tid*3 + k] * task[b*3 + k];
      }
      cpf[tid] = s; taskf[tid] = t2;
    }
    __syncthreads();
    for (int i = tid; i < 192; i += 128)
      h[i] = (i < 128) ? net1[i] : (i < 160 ? cpf[i-128] : taskf[i-160]);
    __syncthreads();
    { float s = b1[tid]; for (int k = 0; k < 192; ++k) s += W1[tid*192+k]*h[k];
      h1[tid] = s > 0.0f ? s : 0.01f * s; }
    __syncthreads();
    { float s = b2[tid]; for (int k = 0; k < 128; ++k) s += W2[tid*128+k]*h1[k];
      h2[tid] = s > 0.0f ? s : 0.01f * s; }
    __syncthreads();
    { float s = b3[tid]; for (int k = 0; k < 128; ++k) s += W3[tid*128+k]*h2[k];
      h3[tid] = s; }
    __syncthreads();
    {
      float sm = bmu[tid], sl = blv[tid];
      for (int k = 0; k < 128; ++k) {
        sm += Wmu[tid*128+k]*h3[k];
        sl += Wlv[tid*128+k]*h3[k];
      }
      zv[tid] = sm + expf(sl * 0.5f) * noise[b*128 + tid];
      out[24 + b*128 + tid]        = sm;   // mu
      out[24 + 1024 + b*128 + tid] = sl;   // logvar
    }
    __syncthreads();
    for (int i = tid; i < 288; i += 128)
      dvec[i] = (i < 128) ? net1[i] : (i < 160 ? cpf[i-128] : zv[i-160]);
    __syncthreads();
    { float s = bd1[tid]; for (int k = 0; k < 288; ++k) s += Wd1[tid*288+k]*dvec[k];
      l1[tid] = s; }
    __syncthreads();
    if (tid < 3) {
      float s = bd2[tid];
      for (int k = 0; k < 128; ++k) s += Wd2[tid*128+k]*l1[k];
      out[b*3 + tid] = s;                  // recon
    }
    __syncthreads();
  }
}

// =====================================================================
// Host orchestration
// =====================================================================
static inline unsigned gblocks(size_t total, int tpb) {
  size_t b = (total + tpb - 1) / tpb;
  if (b > 65536) b = 65536;
  if (b < 1) b = 1;
  return (unsigned)b;
}

extern "C" void kernel_launch(void* const* d_in, const int* in_sizes, int n_in,
                              void* d_out, int out_size, void* d_ws, size_t ws_size,
                              hipStream_t stream) {
  (void)in_sizes; (void)n_in; (void)out_size; (void)ws_size;
  const int B = 8, N = 8192;
  auto F = [&](int i) { return (const float*)d_in[i]; };

  // ---- workspace bump allocator (256B aligned) ----
  char* base = (char*)d_ws;
  size_t off = 0;
  auto allocF = [&](size_t n) {
    float* p = (float*)(base + off);
    off += ((n * sizeof(float) + 255) & ~(size_t)255);
    return p;
  };
  auto allocI = [&](size_t n) {
    int* p = (int*)(base + off);
    off += ((n * sizeof(int) + 255) & ~(size_t)255);
    return p;
  };

  float* xyz0 = allocF((size_t)B*8192*3);  float* feat0 = allocF((size_t)B*8192*4);
  float* xyz1 = allocF((size_t)B*1024*3);  float* feat1 = allocF((size_t)B*1024*64);
  float* xyz2 = allocF((size_t)B*256*3);   float* feat2 = allocF((size_t)B*256*128);
  float* xyz3 = allocF((size_t)B*64*3);    float* feat3 = allocF((size_t)B*64*256);
  float* xyz4 = allocF((size_t)B*16*3);    float* feat4 = allocF((size_t)B*16*512);
  float* uf3  = allocF((size_t)B*64*256);
  float* uf2  = allocF((size_t)B*256*256);
  float* uf1  = allocF((size_t)B*1024*128);
  float* uf0  = allocF((size_t)B*8192*128);
  float* w3b  = allocF((size_t)B*8192*3);
  float* scaleB  = allocF(512);  float* shiftB  = allocF(512);
  float* fcScale = allocF(128);  float* fcShift = allocF(128);
  int* fidxB = allocI((size_t)B*1024);
  int* gidxB = allocI((size_t)B*1024*32);
  int* idx3B = allocI((size_t)B*8192*3);
  float* bufA = allocF(16777216);
  float* bufB = allocF(16777216);

  // conv layer = GEMM -> BN stats -> (optional) fused scale/shift + ReLU
  auto run_conv = [&](const float* X, float* Y, int M, int K, int C, int pW, bool applyRelu) {
    gemm_xwT_kernel<<<dim3(M/64, C/32), 256, 0, stream>>>(X, F(pW), Y, M, K, C);
    float* sc = applyRelu ? scaleB : fcScale;
    float* sh = applyRelu ? shiftB : fcShift;
    bn_stats_kernel<<<C, 256, 0, stream>>>(Y, F(pW+1), F(pW+2), sc, sh, (size_t)M, C);
    if (applyRelu)
      bn_relu_kernel<<<gblocks((size_t)M*C, 256), 256, 0, stream>>>(Y, sc, sh, (size_t)M, C);
  };

  // ---- prep ----
  prep_kernel<<<gblocks((size_t)B*N, 256), 256, 0, stream>>>(F(0), F(1), xyz0, feat0, B, N);

  // ---- set abstraction ----
  auto run_sa = [&](const float* xyzIn, const float* featIn, int Nin, int S,
                    float radius, int ns, int Cf, const int* ch, int nConv,
                    int pbase, float* xyzOut, float* featOut) {
    fps_kernel<<<B, 1024, 0, stream>>>(xyzIn, fidxB, Nin, S);
    gather_xyz_kernel<<<gblocks((size_t)B*S*3, 256), 256, 0, stream>>>(
        xyzIn, fidxB, xyzOut, B, Nin, S);
    ball_query_kernel<<<(B*S + 127)/128, 128, 0, stream>>>(
        xyzOut, xyzIn, gidxB, B, S, Nin, radius*radius, ns);
    group_kernel<<<gblocks((size_t)B*S*ns*(3+Cf), 256), 256, 0, stream>>>(
        xyzIn, featIn, xyzOut, gidxB, bufA, B, S, Nin, ns, Cf);
    const int M = B * S * ns;
    float* cur = bufA;
    for (int j = 0; j < nConv; ++j) {
      float* dst = (cur == bufA) ? bufB : bufA;
      run_conv(cur, dst, M, ch[j], ch[j+1], pbase + 3*j, true);
      cur = dst;
    }
    maxpool_kernel<<<gblocks((size_t)B*S*ch[nConv], 256), 256, 0, stream>>>(
        cur, featOut, B, S, ns, ch[nConv]);
  };

  const int ch0[4] = {7, 32, 32, 64};
  const int ch1[4] = {67, 64, 64, 128};
  const int ch2[4] = {131, 128, 128, 256};
  const int ch3[4] = {259, 256, 256, 512};
  run_sa(xyz0, feat0, 8192, 1024, 0.1f, 32,   4, ch0, 3,  4, xyz1, feat1);
  run_sa(xyz1, feat1, 1024,  256, 0.2f, 32,  64, ch1, 3, 13, xyz2, feat2);
  run_sa(xyz2, feat2,  256,   64, 0.4f, 32, 128, ch2, 3, 22, xyz3, feat3);
  run_sa(xyz3, feat3,   64,   16, 0.8f, 32, 256, ch3, 3, 31, xyz4, feat4);

  // ---- feature propagation (levels 4->3, 3->2, 2->1, 1->0) ----
  auto run_fp = [&](const float* uxyz, const float* kxyz, const float* uf,
                    const float* kf, int Nu, int Nk, int Cu, int Ck,
                    const int* ch, int nConv, int pbase, float* outFeat) {
    three_nn_kernel<<<(B*Nu + 127)/128, 128, 0, stream>>>(uxyz, kxyz, idx3B, w3b, B, Nu, Nk);
    fp_concat_kernel<<<gblocks((size_t)B*Nu*(Ck+Cu), 256), 256, 0, stream>>>(
        idx3B, w3b, kf, uf, bufA, B, Nu, Nk, Ck, Cu);
    const int M = B * Nu;
    float* cur = bufA;
    for (int j = 0; j < nConv; ++j) {
      float* dst = (j == nConv - 1) ? outFeat : ((cur == bufA) ? bufB : bufA);
      run_conv(cur, dst, M, ch[j], ch[j+1], pbase + 3*j, true);
      cur = dst;
    }
  };

  const int fch3[3] = {768, 256, 256};
  const int fch2[3] = {384, 256, 256};
  const int fch1[3] = {320, 256, 128};
  const int fch0[4] = {132, 128, 128, 128};
  run_fp(xyz3, xyz4, feat3, feat4,   64,   16, 256, 512, fch3, 2, 61, uf3);
  run_fp(xyz2, xyz3, feat2, uf3,    256,   64, 128, 256, fch2, 2, 55, uf2);
  run_fp(xyz1, xyz2, feat1, uf2,   1024,  256,  64, 256, fch1, 2, 49, uf1);
  run_fp(xyz0, xyz1, feat0, uf1,   8192, 1024,   4, 128, fch0, 3, 40, uf0);

  // ---- fc layer: full GEMM for BN stats; head applies BN+ReLU to row 0 ----
  run_conv(uf0, bufA, B * N, 128, 128, 67, /*applyRelu=*/false);

  // ---- CVAE head ----
  head_kernel<<<1, 128, 0, stream>>>(
      bufA, fcScale, fcShift, F(1), F(2), F(3),
      F(70), F(71), F(72), F(73),
      F(74), F(75), F(76), F(77), F(78), F(79),
      F(80), F(81), F(82), F(83),
      F(84), F(85), F(86), F(87),
      (float*)d_out, N);
}